// GraphSAGE_57793079935364
// MI455X (gfx1250) — compile-verified
//
#include <hip/hip_runtime.h>
#include <hip/hip_bf16.h>

typedef __attribute__((ext_vector_type(2))) float v2f;
typedef __attribute__((ext_vector_type(8))) float v8f;

#define FEATS 128  // IN_FEATS == H_FEATS == 128

// ---------------------------------------------------------------------------
// Zero helpers
// ---------------------------------------------------------------------------
__global__ void zero_f4_kernel(float4* __restrict__ p, long long n4) {
    long long i = blockIdx.x * (long long)blockDim.x + threadIdx.x;
    long long stride = gridDim.x * (long long)blockDim.x;
    float4 z = make_float4(0.f, 0.f, 0.f, 0.f);
    for (; i < n4; i += stride) p[i] = z;
}

__global__ void zero_f_kernel(float* __restrict__ p, long long n) {
    long long i = blockIdx.x * (long long)blockDim.x + threadIdx.x;
    long long stride = gridDim.x * (long long)blockDim.x;
    for (; i < n; i += stride) p[i] = 0.f;
}

// ---------------------------------------------------------------------------
// Edge scatter: one wave32 per edge. Each lane handles 4 contiguous features
// (float4 gather + 4x global_atomic_add_f32). Lane 0 accumulates in-degree.
// ---------------------------------------------------------------------------
__global__ void sage_scatter_kernel(const float* __restrict__ H,
                                    const int* __restrict__ src,
                                    const int* __restrict__ dst,
                                    float* __restrict__ agg,
                                    float* __restrict__ deg,
                                    int n_edges, int count_deg) {
    int gid  = blockIdx.x * blockDim.x + threadIdx.x;
    int edge = gid >> 5;
    int lane = gid & 31;
    if (edge >= n_edges) return;
    int s = src[edge];
    int d = dst[edge];
    const float4 v = *(const float4*)(H + (size_t)s * FEATS + lane * 4);
    float* out = agg + (size_t)d * FEATS + lane * 4;
    atomicAdd(out + 0, v.x);
    atomicAdd(out + 1, v.y);
    atomicAdd(out + 2, v.z);
    atomicAdd(out + 3, v.w);
    if (count_deg && lane == 0) atomicAdd(deg + d, 1.0f);
}

// ---------------------------------------------------------------------------
// Fused SAGE layer GEMM:
//   Out[16 rows x NCOLS] = act( [A0 | agg/deg] (16x256) @ [Wself;Wneigh] + b )
// One block per 16-row tile. A tile staged in LDS (deg-normalized during
// stage). Each wave computes one 16x16 output tile with v_wmma_f32_16x16x4_f32
// over K=256 (64 WMMA ops). blockDim.x must be (NCOLS/16)*32.
// ---------------------------------------------------------------------------
template <int NCOLS, bool RELU>
__global__ void sage_gemm_kernel(const float* __restrict__ A0,     // [n,128]
                                 const float* __restrict__ Agg,    // [n,128]
                                 const float* __restrict__ deg,    // [n]
                                 const float* __restrict__ Wself,  // [128,NCOLS]
                                 const float* __restrict__ Wneigh, // [128,NCOLS]
                                 const float* __restrict__ bias,   // [NCOLS]
                                 float* __restrict__ Out,          // [n,NCOLS]
                                 int n) {
    // LDS A tile: 16 rows x 256 K, padded row stride 260 floats
    // (260 % 64 == 4 -> 32 lanes of the A-fragment read land on distinct banks)
    __shared__ float ldsA[16][260];

    const int tid  = threadIdx.x;
    const int wave = tid >> 5;
    const int lane = tid & 31;
    const int half = lane >> 4;   // 0: lanes 0-15, 1: lanes 16-31
    const int r    = lane & 15;
    const int row0 = blockIdx.x * 16;

    // --- stage A tile: 16 rows x 64 float4 slots each ---
    for (int idx = tid; idx < 16 * 64; idx += blockDim.x) {
        int rr   = idx >> 6;         // row in tile
        int c4   = (idx & 63) << 2;  // starting col (0..252, step 4)
        int grow = row0 + rr;
        float4 v = make_float4(0.f, 0.f, 0.f, 0.f);
        if (grow < n) {
            if (c4 < FEATS) {
                v = *(const float4*)(A0 + (size_t)grow * FEATS + c4);
            } else {
                float inv = 1.0f / fmaxf(deg[grow], 1.0f);
                float4 t = *(const float4*)(Agg + (size_t)grow * FEATS + (c4 - FEATS));
                v = make_float4(t.x * inv, t.y * inv, t.z * inv, t.w * inv);
            }
        }
        *(float4*)(&ldsA[rr][c4]) = v;
    }
    __syncthreads();

    const int col0 = wave * 16;
    v8f c = {};

    // --- K-loop, self-weight half (K = 0..127) ---
#pragma unroll 4
    for (int k = 0; k < FEATS; k += 4) {
        int kb = k + 2 * half;
        v2f a = *(const v2f*)(&ldsA[r][kb]);  // A[M=r][K=kb..kb+1]
        v2f b;
        b.x = Wself[(size_t)kb * NCOLS + col0 + r];        // B[K=kb][N=col0+r]
        b.y = Wself[(size_t)(kb + 1) * NCOLS + col0 + r];
        c = __builtin_amdgcn_wmma_f32_16x16x4_f32(false, a, false, b,
                                                  (short)0, c, false, false);
    }
    // --- K-loop, neighbor-weight half (K = 128..255) ---
#pragma unroll 4
    for (int k = 0; k < FEATS; k += 4) {
        int kb = k + 2 * half;
        v2f a = *(const v2f*)(&ldsA[r][FEATS + kb]);
        v2f b;
        b.x = Wneigh[(size_t)kb * NCOLS + col0 + r];
        b.y = Wneigh[(size_t)(kb + 1) * NCOLS + col0 + r];
        c = __builtin_amdgcn_wmma_f32_16x16x4_f32(false, a, false, b,
                                                  (short)0, c, false, false);
    }

    // --- epilogue: bias (+ReLU), D element [row0 + v + 8*half][col0 + r] ---
    float bcol = bias[col0 + r];
#pragma unroll
    for (int v = 0; v < 8; ++v) {
        int grow = row0 + v + 8 * half;
        if (grow < n) {
            float val = c[v] + bcol;
            if (RELU) val = fmaxf(val, 0.0f);
            Out[(size_t)grow * NCOLS + col0 + r] = val;
        }
    }
}

// ---------------------------------------------------------------------------
// Launch
// ---------------------------------------------------------------------------
extern "C" void kernel_launch(void* const* d_in, const int* in_sizes, int n_in,
                              void* d_out, int out_size, void* d_ws, size_t ws_size,
                              hipStream_t stream) {
    const float* x       = (const float*)d_in[0];
    const int*   src     = (const int*)d_in[1];
    const int*   dst     = (const int*)d_in[2];
    const float* W_self1 = (const float*)d_in[3];
    const float* W_neigh1= (const float*)d_in[4];
    const float* b1      = (const float*)d_in[5];
    const float* W_self2 = (const float*)d_in[6];
    const float* W_neigh2= (const float*)d_in[7];
    const float* b2      = (const float*)d_in[8];
    float*       out     = (float*)d_out;

    const int n_nodes = in_sizes[0] / FEATS;
    const int n_edges = in_sizes[1];

    // workspace layout (floats): agg[n*128] | deg[n] | h1[n*128]
    float* agg = (float*)d_ws;
    float* deg = agg + (size_t)n_nodes * FEATS;
    float* h1  = deg + (size_t)n_nodes;

    const long long aggN4 = ((long long)n_nodes * FEATS) / 4;

    dim3 zgrid(2048), zblk(256);
    dim3 sgrid((unsigned)((((long long)n_edges * 32) + 255) / 256)), sblk(256);
    dim3 ggrid((unsigned)((n_nodes + 15) / 16));

    // ---- layer 1 ----
    zero_f4_kernel<<<zgrid, zblk, 0, stream>>>((float4*)agg, aggN4);
    zero_f_kernel <<<zgrid, zblk, 0, stream>>>(deg, (long long)n_nodes);
    sage_scatter_kernel<<<sgrid, sblk, 0, stream>>>(x, src, dst, agg, deg,
                                                    n_edges, /*count_deg=*/1);
    sage_gemm_kernel<128, true><<<ggrid, dim3(256), 0, stream>>>(
        x, agg, deg, W_self1, W_neigh1, b1, h1, n_nodes);

    // ---- layer 2 (reuse agg buffer; deg is unchanged) ----
    zero_f4_kernel<<<zgrid, zblk, 0, stream>>>((float4*)agg, aggN4);
    sage_scatter_kernel<<<sgrid, sblk, 0, stream>>>(h1, src, dst, agg, deg,
                                                    n_edges, /*count_deg=*/0);
    sage_gemm_kernel<64, false><<<ggrid, dim3(128), 0, stream>>>(
        h1, agg, deg, W_self2, W_neigh2, b2, out, n_nodes);
}